// KalmanGraphicalModel_60679297958187
// MI455X (gfx1250) — compile-verified
//
#include <hip/hip_runtime.h>

// ---------------------------------------------------------------------------
// Kalman graphical-model smoother, CDNA5 (gfx1250) implementation.
//
// Each iteration is the linear 3-point stencil
//   x_new[:,t] = A x[:,t-1] + B x[:,t] + C x[:,t+1] + D y[:,t]
// realized as V_WMMA_F32_16X16X4_F32 chains:
//   D(16x16) = Gs(16x56) * V(56x16),  Gs = blockdiag(G8, G8), G8 = [A|B|C|D]
// One wave32 handles 32 trajectory columns per tile, several tiles per wave
// (grid-stride) with the Gs operand registers hoisted out of the tile loop.
// All indexing is 32-bit so global loads use the saddr+voffset form.
// ---------------------------------------------------------------------------

typedef __attribute__((ext_vector_type(2))) float v2f;
typedef __attribute__((ext_vector_type(8))) float v8f;

#define KG_N 8
#define KG_M 4
#define KG_ITERS 100
#define KG_TILES_PER_WAVE 4

// ---------------- setup: build the fused stencil operator Gs (16x56) --------

__device__ __forceinline__ void kg_invert(const float* src, float* inv, int n) {
    // Gauss-Jordan with partial pivoting, n <= 8. Single-thread use only.
    float a[8][16];
    for (int i = 0; i < n; ++i)
        for (int j = 0; j < n; ++j) {
            a[i][j]     = src[i * n + j];
            a[i][n + j] = (i == j) ? 1.0f : 0.0f;
        }
    for (int c = 0; c < n; ++c) {
        int p = c;
        float best = fabsf(a[c][c]);
        for (int r = c + 1; r < n; ++r) {
            float v = fabsf(a[r][c]);
            if (v > best) { best = v; p = r; }
        }
        if (p != c)
            for (int j = 0; j < 2 * n; ++j) {
                float tmp = a[c][j]; a[c][j] = a[p][j]; a[p][j] = tmp;
            }
        float d = 1.0f / a[c][c];
        for (int j = 0; j < 2 * n; ++j) a[c][j] *= d;
        for (int r = 0; r < n; ++r) {
            if (r == c) continue;
            float f = a[r][c];
            for (int j = 0; j < 2 * n; ++j) a[r][j] -= f * a[c][j];
        }
    }
    for (int i = 0; i < n; ++i)
        for (int j = 0; j < n; ++j) inv[i * n + j] = a[i][n + j];
}

__global__ void kg_setup(const float* __restrict__ F, const float* __restrict__ Hm,
                         const float* __restrict__ Q, const float* __restrict__ R,
                         const float* __restrict__ gamma_p, float* __restrict__ Gs) {
    if (threadIdx.x != 0 || blockIdx.x != 0) return;
    const float g = gamma_p[0];

    float Qi[KG_N * KG_N], Ri[KG_M * KG_M];
    kg_invert(Q, Qi, KG_N);
    kg_invert(R, Ri, KG_M);

    // FtQi = F^T Qinv (8x8);  HtRi = H^T Rinv (8x4)
    float FtQi[KG_N * KG_N], HtRi[KG_N * KG_M];
    for (int i = 0; i < KG_N; ++i)
        for (int j = 0; j < KG_N; ++j) {
            float s = 0.0f;
            for (int k = 0; k < KG_N; ++k) s += F[k * KG_N + i] * Qi[k * KG_N + j];
            FtQi[i * KG_N + j] = s;
        }
    for (int i = 0; i < KG_N; ++i)
        for (int j = 0; j < KG_M; ++j) {
            float s = 0.0f;
            for (int k = 0; k < KG_M; ++k) s += Hm[k * KG_N + i] * Ri[k * KG_M + j];
            HtRi[i * KG_M + j] = s;
        }

    // Row-zeroing of d1/d2 folds into column-zeroing of the operators:
    // P1 = (-Qinv) with column 0 zeroed; P2 = FtQinv with column 7 zeroed.
    float P1[KG_N * KG_N], P2[KG_N * KG_N];
    for (int i = 0; i < KG_N; ++i)
        for (int j = 0; j < KG_N; ++j) {
            P1[i * KG_N + j] = (j == 0) ? 0.0f : -Qi[i * KG_N + j];
            P2[i * KG_N + j] = (j == KG_N - 1) ? 0.0f : FtQi[i * KG_N + j];
        }

    // Cpast = -P1 F ; Ccur = P1 - P2 F - HtRi H ; Cfut = P2 ; Cy = HtRi
    float Cpast[KG_N * KG_N], Ccur[KG_N * KG_N];
    for (int i = 0; i < KG_N; ++i)
        for (int j = 0; j < KG_N; ++j) {
            float s1 = 0.0f, s2 = 0.0f, s3 = 0.0f;
            for (int k = 0; k < KG_N; ++k) {
                s1 += P1[i * KG_N + k] * F[k * KG_N + j];
                s2 += P2[i * KG_N + k] * F[k * KG_N + j];
            }
            for (int k = 0; k < KG_M; ++k) s3 += HtRi[i * KG_M + k] * Hm[k * KG_N + j];
            Cpast[i * KG_N + j] = -s1;
            Ccur[i * KG_N + j]  = P1[i * KG_N + j] - s2 - s3;
        }

    // G8 (8x28) = [g*Cpast | I + g*Ccur | g*Cfut | g*Cy]
    float G8[KG_N * 28];
    for (int i = 0; i < KG_N; ++i) {
        for (int j = 0; j < KG_N; ++j) {
            G8[i * 28 + j]      = g * Cpast[i * KG_N + j];
            G8[i * 28 + 8 + j]  = g * Ccur[i * KG_N + j] + ((i == j) ? 1.0f : 0.0f);
            G8[i * 28 + 16 + j] = g * P2[i * KG_N + j];
        }
        for (int j = 0; j < KG_M; ++j) G8[i * 28 + 24 + j] = g * HtRi[i * KG_M + j];
    }

    // Gs (16x56) = blockdiag(G8, G8)
    for (int i = 0; i < 16 * 56; ++i) Gs[i] = 0.0f;
    for (int i = 0; i < KG_N; ++i)
        for (int j = 0; j < 28; ++j) {
            Gs[i * 56 + j]            = G8[i * 28 + j];
            Gs[(i + 8) * 56 + 28 + j] = G8[i * 28 + j];
        }
}

// ---------------- one stencil iteration via WMMA ---------------------------

__device__ __forceinline__ float kg_fetch(const float* __restrict__ x,
                                          const float* __restrict__ ys,
                                          int T, int kk, int t, int tm, int tp) {
    // V[kk, t] of the 28-row per-column input vector [x(t-1); x(t); x(t+1); y(t)]
    if (kk < 8)  return x[kk * T + tm];
    if (kk < 16) return x[(kk - 8) * T + t];
    if (kk < 24) return x[(kk - 16) * T + tp];
    return ys[(kk - 24) * T + t];
}

__global__ __launch_bounds__(256) void kg_step(const float* __restrict__ x,
                                               const float* __restrict__ ys,
                                               const float* __restrict__ Gs,
                                               float* __restrict__ xo, int T) {
    const int lane   = threadIdx.x & 31;
    const int wid    = blockIdx.x * (blockDim.x >> 5) + (threadIdx.x >> 5);
    const int nwaves = gridDim.x * (blockDim.x >> 5);
    const int col    = lane & 15;
    const int half   = lane >> 4;

    // A operand (hoisted, reused for every tile): lane row = col;
    // lanes 0-15 supply K = k0,k0+1; lanes 16-31 supply K = k0+2,k0+3.
    float ga[14], gb[14];
    const float* Gb = Gs + col * 56 + half * 2;
#pragma unroll
    for (int c = 0; c < 14; ++c) {
        ga[c] = Gb[4 * c];
        gb[c] = Gb[4 * c + 1];
    }

    // Grid-stride over 32-column tiles; loop bounds are wave-uniform so EXEC
    // is all-ones at every WMMA.
    for (int t0 = wid * 32; t0 < T; t0 += nwaves * 32) {
        // Tile-A columns t0+col, tile-B columns t0+16+col (clamped taps)
        int tA  = min(t0 + col, T - 1);
        int tAm = max(tA - 1, 0);
        int tAp = min(tA + 1, T - 1);
        int tB  = min(t0 + 16 + col, T - 1);
        int tBm = tB - 1;                  // tB >= 16, never clamps at 0
        int tBp = min(tB + 1, T - 1);

        float va[14], vb[14];
#pragma unroll
        for (int c = 0; c < 14; ++c) {
            const int k0 = 4 * c + 2 * half;
            const int kk = (c < 7) ? k0 : (k0 - 28);
            const int t  = (c < 7) ? tA  : tB;
            const int tm = (c < 7) ? tAm : tBm;
            const int tp = (c < 7) ? tAp : tBp;
            va[c] = kg_fetch(x, ys, T, kk,     t, tm, tp);
            vb[c] = kg_fetch(x, ys, T, kk + 1, t, tm, tp);
        }

        v8f acc = {};
#pragma unroll
        for (int c = 0; c < 14; ++c) {
            v2f a = {ga[c], gb[c]};
            v2f b = {va[c], vb[c]};
            acc = __builtin_amdgcn_wmma_f32_16x16x4_f32(
                /*neg_a=*/false, a, /*neg_b=*/false, b,
                /*c_mod=*/(short)0, acc, /*reuse_a=*/false, /*reuse_b=*/false);
        }

        // D layout: VGPR v -> (M=v, N=lane) lanes 0-15, (M=v+8, N=lane-16)
        // lanes 16-31. With Gs block-diagonal both collapse to column
        // t0+lane, row v: fully coalesced 128B stores.
        const int ts = t0 + lane;
        if (ts < T) {
#pragma unroll
            for (int v = 0; v < 8; ++v) xo[v * T + ts] = acc[v];
        }
    }
}

// ---------------- launch ---------------------------------------------------

extern "C" void kernel_launch(void* const* d_in, const int* in_sizes, int n_in,
                              void* d_out, int out_size, void* d_ws, size_t ws_size,
                              hipStream_t stream) {
    const float* xs    = (const float*)d_in[0];
    const float* ys    = (const float*)d_in[1];
    const float* F     = (const float*)d_in[2];
    const float* Hm    = (const float*)d_in[3];
    const float* Q     = (const float*)d_in[4];
    const float* R     = (const float*)d_in[5];
    const float* gamma = (const float*)d_in[6];
    float* out = (float*)d_out;

    const int T = in_sizes[0] / KG_N;  // xs is N x T

    float* Gs   = (float*)d_ws;                  // 16*56*4 = 3584 B
    float* bufA = (float*)((char*)d_ws + 4096);  // N*T floats (~16 MB)

    kg_setup<<<1, 1, 0, stream>>>(F, Hm, Q, R, gamma, Gs);

    const int tiles  = (T + 31) / 32;
    const int waves  = (tiles + KG_TILES_PER_WAVE - 1) / KG_TILES_PER_WAVE;
    const int blocks = (waves + 7) / 8;  // 256 threads = 8 wave32 per block

    // it 0: xs -> bufA; odd its: bufA -> out; even its >= 2: out -> bufA.
    // KG_ITERS = 100 => iteration 99 (odd) writes d_out. No final copy, no
    // input mutation, fully deterministic.
    for (int it = 0; it < KG_ITERS; ++it) {
        const float* src = (it == 0) ? xs : ((it & 1) ? bufA : out);
        float* dst       = (it & 1) ? out : bufA;
        kg_step<<<blocks, 256, 0, stream>>>(src, ys, Gs, dst, T);
    }
}